// Caps_Layer_5566277616445
// MI455X (gfx1250) — compile-verified
//
#include <hip/hip_runtime.h>

typedef float v2f __attribute__((ext_vector_type(2)));
typedef float v8f __attribute__((ext_vector_type(8)));

#define NUM_CAP 16
#define DIM_CAP 64
#define B_SZ    64
#define IN_SZ   512
#define K_SZ    768
#define N_SZ    1024                 // NUM_CAP * DIM_CAP
#define M_SZ    (B_SZ * IN_SZ)       // 32768

#define BM 128
#define BN 128
#define BK 16
#define PITCH 288   // 2*128 interleaved floats + 32 pad -> +32 banks per kpair row

// ---------------------------------------------------------------------------
// GEMM via V_WMMA_F32_16X16X4_F32.
// LDS tiles stored K-pair-interleaved: tile[kpair][col][2], so each WMMA
// fragment is ONE ds_load_b64 (no operand shuffling). Double-buffered LDS,
// global prefetch into registers overlaps the WMMA block.
// Block: 256 threads = 8 waves in 4(M) x 2(N); wave tile 32M x 64N.
// ---------------------------------------------------------------------------
__global__ __launch_bounds__(256)
void caps_gemm_wmma(const float* __restrict__ x,
                    const float* __restrict__ W,
                    float* __restrict__ u)
{
    __shared__ float As2[2][8 * PITCH];   // [buf][kpair][m][2]
    __shared__ float Bs2[2][8 * PITCH];   // [buf][kpair][n][2]

    const int tid  = threadIdx.x;
    const int wave = tid >> 5;
    const int lane = tid & 31;
    const int hh   = lane >> 4;          // lane half: selects kpair row +0/+1
    const int lm   = lane & 15;
    const int wm   = wave & 3;           // M group (32 rows)
    const int wn   = wave >> 2;          // N group (64 cols)

    const int m0 = blockIdx.x * BM;
    const int n0 = blockIdx.y * BN;

    v8f zero = {};
    v8f acc[2][4];
#pragma unroll
    for (int s = 0; s < 2; ++s)
#pragma unroll
        for (int t = 0; t < 4; ++t) acc[s][t] = zero;

    float4 a_pre[2];
    float2 b0_pre[2], b1_pre[2];

    auto load_tile = [&](int k0) {
#pragma unroll
        for (int i = 0; i < 2; ++i) {
            const int c  = tid + i * 256;
            const int mi = c >> 2, kq = c & 3;              // A: 128 m x 4 k-quads
            a_pre[i] = *(const float4*)(x + (size_t)(m0 + mi) * K_SZ + k0 + kq * 4);

            const int kp = c >> 6, np = c & 63;             // B: 8 kpairs x 64 n-pairs
            const float* wp = W + (size_t)(k0 + 2 * kp) * N_SZ + n0 + np * 2;
            b0_pre[i] = *(const float2*)(wp);
            b1_pre[i] = *(const float2*)(wp + N_SZ);
        }
    };

    auto store_tile = [&](int buf) {
#pragma unroll
        for (int i = 0; i < 2; ++i) {
            const int c  = tid + i * 256;
            const int mi = c >> 2, kq = c & 3;
            *(float2*)(&As2[buf][(2 * kq + 0) * PITCH + mi * 2]) =
                make_float2(a_pre[i].x, a_pre[i].y);
            *(float2*)(&As2[buf][(2 * kq + 1) * PITCH + mi * 2]) =
                make_float2(a_pre[i].z, a_pre[i].w);

            const int kp = c >> 6, np = c & 63;             // n = 2*np
            *(float4*)(&Bs2[buf][kp * PITCH + np * 4]) =
                make_float4(b0_pre[i].x, b1_pre[i].x, b0_pre[i].y, b1_pre[i].y);
        }
    };

    auto compute = [&](int buf) {
#pragma unroll
        for (int kk = 0; kk < BK; kk += 4) {
            const int kp = (kk >> 1) + hh;                  // kpair row for this lane half
            const v2f a0 = *(const v2f*)(&As2[buf][kp * PITCH + (wm * 32 +  0 + lm) * 2]);
            const v2f a1 = *(const v2f*)(&As2[buf][kp * PITCH + (wm * 32 + 16 + lm) * 2]);
#pragma unroll
            for (int t = 0; t < 4; ++t) {
                const v2f b = *(const v2f*)(&Bs2[buf][kp * PITCH + (wn * 64 + t * 16 + lm) * 2]);
                acc[0][t] = __builtin_amdgcn_wmma_f32_16x16x4_f32(
                    false, a0, false, b, (short)0, acc[0][t], false, false);
                acc[1][t] = __builtin_amdgcn_wmma_f32_16x16x4_f32(
                    false, a1, false, b, (short)0, acc[1][t], false, false);
            }
        }
    };

    load_tile(0);
    store_tile(0);
    __syncthreads();

    for (int k0 = 0; k0 < K_SZ; k0 += BK) {
        const int buf = (k0 >> 4) & 1;
        const bool more = (k0 + BK) < K_SZ;
        if (more) load_tile(k0 + BK);       // global prefetch overlaps WMMAs below
        compute(buf);
        if (more) store_tile(buf ^ 1);
        __syncthreads();
    }

    // ---- epilogue: D layout reg r: lanes0-15 M=r, lanes16-31 M=r+8; N=lm ----
#pragma unroll
    for (int s = 0; s < 2; ++s) {
#pragma unroll
        for (int t = 0; t < 4; ++t) {
#pragma unroll
            for (int r = 0; r < 8; ++r) {
                const int gm = m0 + wm * 32 + s * 16 + r + 8 * hh;
                const int gn = n0 + wn * 64 + t * 16 + lm;
                const int bb = gm >> 9, j  = gm & 511;       // gm = bb*512 + j
                const int ic = gn >> 6, kd = gn & 63;        // gn = ic*64 + kd
                u[(((size_t)bb * NUM_CAP + ic) * IN_SZ + j) * DIM_CAP + kd] = acc[s][t][r];
            }
        }
    }
}

// ---------------------------------------------------------------------------
// Dynamic routing: one block per batch element, 1024 threads (32 waves).
// b logits + c coefficients live in LDS; u slices stream from L2 (u fits in L2).
// ---------------------------------------------------------------------------
__global__ __launch_bounds__(1024)
void caps_routing(const float* __restrict__ u, float* __restrict__ out)
{
    __shared__ float b_lds[NUM_CAP * IN_SZ];     // 32 KB
    __shared__ float c_lds[NUM_CAP * IN_SZ];     // 32 KB
    __shared__ float s_lds[NUM_CAP * DIM_CAP];   // 4 KB, also holds squashed out
    __shared__ float inv_lds[NUM_CAP];

    const int tid = threadIdx.x;
    const int bb  = blockIdx.x;
    const float* ub = u + (size_t)bb * NUM_CAP * IN_SZ * DIM_CAP;

    for (int p = tid; p < NUM_CAP * IN_SZ; p += 1024) b_lds[p] = 0.0f;
    __syncthreads();

    for (int it = 0; it < 3; ++it) {
        // softmax over capsule axis (16) per input position j
        if (tid < IN_SZ) {
            const int j = tid;
            float m = -3.402823466e38f;
#pragma unroll
            for (int i = 0; i < NUM_CAP; ++i) m = fmaxf(m, b_lds[i * IN_SZ + j]);
            float e[NUM_CAP];
            float den = 0.0f;
#pragma unroll
            for (int i = 0; i < NUM_CAP; ++i) {
                e[i] = __expf(b_lds[i * IN_SZ + j] - m);
                den += e[i];
            }
            const float rden = 1.0f / den;
#pragma unroll
            for (int i = 0; i < NUM_CAP; ++i) c_lds[i * IN_SZ + j] = e[i] * rden;
        }
        __syncthreads();

        // s[i][k] = sum_j c[i][j] * u[i][j][k]   (thread = (i,k))
        {
            const int i = tid >> 6;
            const int k = tid & 63;
            const float* up = ub + ((size_t)i * IN_SZ) * DIM_CAP + k;
            const float* cp = c_lds + i * IN_SZ;
            float acc = 0.0f;
            for (int j = 0; j < IN_SZ; ++j)
                acc = fmaf(cp[j], up[(size_t)j * DIM_CAP], acc);
            s_lds[tid] = acc;
        }
        __syncthreads();

        if (tid < NUM_CAP) {                     // squash scale per capsule
            float ss = 0.0f;
#pragma unroll
            for (int k = 0; k < DIM_CAP; ++k) {
                const float v = s_lds[tid * DIM_CAP + k];
                ss = fmaf(v, v, ss);
            }
            inv_lds[tid] = 1.0f / sqrtf(ss + 1e-7f);
        }
        __syncthreads();
        s_lds[tid] *= inv_lds[tid >> 6];
        __syncthreads();

        if (it < 2) {
            // b[i][j] = sum_k out[i][k] * u[i][j][k]  (8192 dots / 1024 threads)
#pragma unroll
            for (int p = 0; p < 8; ++p) {
                const int idx = tid + p * 1024;
                const int i = idx >> 9;
                const int j = idx & 511;
                const float4* up4 = (const float4*)(ub + ((size_t)i * IN_SZ + j) * DIM_CAP);
                const float4* op4 = (const float4*)(s_lds + i * DIM_CAP);
                float acc = 0.0f;
#pragma unroll
                for (int q = 0; q < 16; ++q) {
                    const float4 uu = up4[q];
                    const float4 oo = op4[q];
                    acc = fmaf(uu.x, oo.x, acc);
                    acc = fmaf(uu.y, oo.y, acc);
                    acc = fmaf(uu.z, oo.z, acc);
                    acc = fmaf(uu.w, oo.w, acc);
                }
                b_lds[idx] = acc;
            }
            __syncthreads();
        }
    }

    // out[b, i, k]; tid == i*64 + k
    out[(size_t)bb * NUM_CAP * DIM_CAP + tid] = s_lds[tid];
}

extern "C" void kernel_launch(void* const* d_in, const int* in_sizes, int n_in,
                              void* d_out, int out_size, void* d_ws, size_t ws_size,
                              hipStream_t stream)
{
    const float* x = (const float*)d_in[0];   // [64, 512, 768]
    const float* W = (const float*)d_in[1];   // [1, 768, 1024]
    float* u   = (float*)d_ws;                // [64, 16, 512, 64] = 128 MB scratch
    float* out = (float*)d_out;               // [64, 16, 64]

    dim3 gemm_grid(M_SZ / BM, N_SZ / BN);     // 256 x 8 workgroups
    caps_gemm_wmma<<<gemm_grid, 256, 0, stream>>>(x, W, u);
    caps_routing<<<B_SZ, 1024, 0, stream>>>(u, out);
}